// RET_25503515804202
// MI455X (gfx1250) — compile-verified
//
#include <hip/hip_runtime.h>
#include <hip/hip_bf16.h>

// ---- problem constants (match reference) ----
#define LVV 2048   // video length
#define LTT 512    // text length
#define LL  2560   // LVV + LTT
#define DD  512    // d_model
#define HH  8      // heads
#define DKK 64     // head dim
#define FF  1024   // d_ff
#define LOGGAMMA (-0.10536051565782630f)  // log(0.9)

typedef float v2f __attribute__((ext_vector_type(2)));
typedef float v8f __attribute__((ext_vector_type(8)));

// =====================================================================
// f32 WMMA GEMM:  C[m,n] = epilogue( scale*(sum_k A[m,k]*B(k,n) + bias[n]) )
//   A: row-major M x K, row stride lda, column offset aoff
//   B: bKN==0 -> stored N x K row-major (weights; contraction over inner dim)
//      bKN==1 -> stored K x N row-major (e.g. V matrix, P @ V)
// Each wave computes a 32x32 C macro-tile: 2 A-frags x 2 B-frags -> 4
// v_wmma_f32_16x16x4_f32 per k-step (wmma:vmem 1:1, 8 FLOP/B per lane).
// Block = 4 waves (2x2) -> 64x64 tile. All dims here are multiples of 64.
// f32 16x16x4 lane layout (ISA 7.12.2): A/B: lane%16 = row, lane/16 selects
// k-pair {k,k+1}/{k+2,k+3}; C: vgpr j: lane<16 -> row j, lane>=16 -> row j+8.
// =====================================================================
__global__ __launch_bounds__(128) void gemm_wmma_f32(
    const float* __restrict__ A, int lda, int aoff,
    const float* __restrict__ B, int ldb, int boff, int bKN,
    float* __restrict__ C, int ldc, int coff,
    const float* __restrict__ bias,
    const float* __restrict__ res, int ldr, int roff,
    float scale, int act, int K)
{
  const int lane = threadIdx.x & 31;
  const int wave = threadIdx.x >> 5;
  const int m0 = blockIdx.y * 64 + (wave >> 1) * 32;
  const int n0 = blockIdx.x * 64 + (wave & 1) * 32;
  const int r  = lane & 15;          // row within a 16-row fragment
  const int kp = (lane >> 4) << 1;   // 0 or 2: k-pair held by this half-wave

  v8f c00 = {0.f,0.f,0.f,0.f,0.f,0.f,0.f,0.f};
  v8f c01 = c00, c10 = c00, c11 = c00;

  const float* A0 = A + (size_t)(m0 + r) * lda + aoff + kp;
  const float* A1 = A0 + (size_t)16 * lda;

  if (bKN == 0) {
    const float* B0 = B + (size_t)(n0 + r) * ldb + boff + kp;
    const float* B1 = B0 + (size_t)16 * ldb;
#pragma unroll 2
    for (int k = 0; k < K; k += 4) {
      if ((k & 31) == 0 && k + 32 < K) {   // uniform: cacheline-ahead prefetch
        __builtin_prefetch(A0 + k + 32, 0, 1);
        __builtin_prefetch(A1 + k + 32, 0, 1);
        __builtin_prefetch(B0 + k + 32, 0, 1);
        __builtin_prefetch(B1 + k + 32, 0, 1);
      }
      v2f a0 = *(const v2f*)(A0 + k);
      v2f a1 = *(const v2f*)(A1 + k);
      v2f b0 = *(const v2f*)(B0 + k);
      v2f b1 = *(const v2f*)(B1 + k);
      c00 = __builtin_amdgcn_wmma_f32_16x16x4_f32(false, a0, false, b0, (short)0, c00, false, false);
      c01 = __builtin_amdgcn_wmma_f32_16x16x4_f32(false, a0, false, b1, (short)0, c01, false, false);
      c10 = __builtin_amdgcn_wmma_f32_16x16x4_f32(false, a1, false, b0, (short)0, c10, false, false);
      c11 = __builtin_amdgcn_wmma_f32_16x16x4_f32(false, a1, false, b1, (short)0, c11, false, false);
    }
  } else {
    // B stored K x N: column pointers bumped by 4*ldb per step (no per-k mul)
    const float* Bc0 = B + (size_t)kp * ldb + boff + n0 + r;
    const float* Bc1 = Bc0 + 16;
    const size_t step = (size_t)4 * ldb;
#pragma unroll 2
    for (int k = 0; k < K; k += 4) {
      v2f a0 = *(const v2f*)(A0 + k);
      v2f a1 = *(const v2f*)(A1 + k);
      v2f b0, b1;
      b0.x = Bc0[0]; b0.y = Bc0[ldb];
      b1.x = Bc1[0]; b1.y = Bc1[ldb];
      Bc0 += step; Bc1 += step;
      c00 = __builtin_amdgcn_wmma_f32_16x16x4_f32(false, a0, false, b0, (short)0, c00, false, false);
      c01 = __builtin_amdgcn_wmma_f32_16x16x4_f32(false, a0, false, b1, (short)0, c01, false, false);
      c10 = __builtin_amdgcn_wmma_f32_16x16x4_f32(false, a1, false, b0, (short)0, c10, false, false);
      c11 = __builtin_amdgcn_wmma_f32_16x16x4_f32(false, a1, false, b1, (short)0, c11, false, false);
    }
  }

  const int half = lane >> 4;
  const int cc   = lane & 15;
#define EPILOGUE(ACC, MOFF, NOFF)                                           \
  {                                                                         \
    _Pragma("unroll")                                                       \
    for (int j = 0; j < 8; ++j) {                                           \
      const int row = m0 + (MOFF) + j + half * 8;                           \
      const int col = n0 + (NOFF) + cc;                                     \
      float v = (ACC)[j];                                                   \
      if (bias) v += bias[col];                                             \
      v *= scale;                                                           \
      if (act == 1) v = 0.5f * v * (1.f + erff(v * 0.70710678118654752f));  \
      if (res) v += res[(size_t)row * ldr + roff + col];                    \
      C[(size_t)row * ldc + coff + col] = v;                                \
    }                                                                       \
  }
  EPILOGUE(c00, 0, 0)
  EPILOGUE(c01, 0, 16)
  EPILOGUE(c10, 16, 0)
  EPILOGUE(c11, 16, 16)
#undef EPILOGUE
}

// ---- concat [src_vid ; src_txt] -> srcbuf (LL x DD) ----
__global__ __launch_bounds__(256) void concat_kernel(
    const float* __restrict__ vid, const float* __restrict__ txt,
    float* __restrict__ dst)
{
  const int row = blockIdx.x, t = threadIdx.x;
  const float* s = (row < LVV) ? vid + (size_t)row * DD
                               : txt + (size_t)(row - LVV) * DD;
  float* d = dst + (size_t)row * DD;
  d[t]       = s[t];
  d[t + 256] = s[t + 256];
}

// ---- xPos rotary positional embedding of src_vid (2048 x 512) ----
__global__ __launch_bounds__(256) void xpos_kernel(
    const float* __restrict__ x, float* __restrict__ out, int downscale)
{
  const int n = blockIdx.x;       // position
  const int i = threadIdx.x;      // pair index 0..255
  const float s  = (2.f * i + 0.4f * DD) / (1.4f * DD);
  float sc = expf(((float)n / 512.f) * logf(s));
  if (downscale) sc = 1.f / sc;
  const float invf = expf(-((float)i / 256.f) * logf(10000.f));
  const float ang  = (float)n * invf;
  const float sn = sinf(ang) * sc, cs = cosf(ang) * sc;
  const float x1 = x[(size_t)n * DD + 2 * i];
  const float x2 = x[(size_t)n * DD + 2 * i + 1];
  out[(size_t)n * DD + 2 * i]     = x1 * cs - x2 * sn;
  out[(size_t)n * DD + 2 * i + 1] = x2 * cs + x1 * sn;
}

// ---- sine positional embedding for text (512 x 512) ----
__global__ __launch_bounds__(256) void sinepos_kernel(float* __restrict__ out)
{
  const int l = blockIdx.x, i = threadIdx.x;
  const float xe = ((float)(l + 1)) / (512.f + 1e-6f) * 6.283185307179586f;
  const float a  = xe * expf(-((float)i / 256.f) * logf(10000.f));
  out[(size_t)l * DD + 2 * i]     = sinf(a);
  out[(size_t)l * DD + 2 * i + 1] = cosf(a);
}

// ---- LayerNorm over rows of length 512 (fixed-order tree reduce) ----
__global__ __launch_bounds__(256) void ln_kernel(
    const float* __restrict__ in, const float* __restrict__ w,
    const float* __restrict__ b, float* __restrict__ out)
{
  const int row = blockIdx.x, t = threadIdx.x;
  const float* p = in + (size_t)row * DD;
  const float x0 = p[t], x1 = p[t + 256];
  __shared__ float rs[256], rq[256];
  rs[t] = x0 + x1;
  rq[t] = x0 * x0 + x1 * x1;
  __syncthreads();
  for (int st = 128; st > 0; st >>= 1) {
    if (t < st) { rs[t] += rs[t + st]; rq[t] += rq[t + st]; }
    __syncthreads();
  }
  const float mean = rs[0] * (1.f / DD);
  const float var  = rq[0] * (1.f / DD) - mean * mean;
  const float inv  = rsqrtf(var + 1e-5f);
  float* o = out + (size_t)row * DD;
  o[t]       = (x0 - mean) * inv * w[t] + b[t];
  o[t + 256] = (x1 - mean) * inv * w[t + 256] + b[t + 256];
}

// ---- q = src2_vid + posq ; k = src2_vid + posk ----
__global__ __launch_bounds__(256) void qk_build_kernel(
    const float* __restrict__ src2, const float* __restrict__ posq,
    const float* __restrict__ posk, float* __restrict__ q, float* __restrict__ k)
{
  const int row = blockIdx.x, t = threadIdx.x;
  const size_t base = (size_t)row * DD;
#pragma unroll
  for (int e = t; e < DD; e += 256) {
    const float s = src2[base + e];
    q[base + e] = s + posq[base + e];
    k[base + e] = s + posk[base + e];
  }
}

// ---- fused retention-mask multiply + row softmax, in place on S (2048 x 2048) ----
__global__ __launch_bounds__(256) void mask_softmax_kernel(
    float* __restrict__ S, const float* __restrict__ lw)
{
  const int l = blockIdx.x, t = threadIdx.x;
  float x[8];
#pragma unroll
  for (int i = 0; i < 8; ++i) {
    const int j = t + i * 256;
    float m;
    if (j > l)      m = 0.f;
    else if (j == l) m = 1.f;
    else {
      const float sig = 1.f / (1.f + expf(-lw[(size_t)l * LVV + j]));
      m = expf((float)(l - j) * LOGGAMMA) * sig;
    }
    x[i] = S[(size_t)l * LVV + j] * m;
  }
  __shared__ float red[256];
  float mx = x[0];
#pragma unroll
  for (int i = 1; i < 8; ++i) mx = fmaxf(mx, x[i]);
  red[t] = mx; __syncthreads();
  for (int st = 128; st > 0; st >>= 1) {
    if (t < st) red[t] = fmaxf(red[t], red[t + st]);
    __syncthreads();
  }
  mx = red[0]; __syncthreads();
  float sm = 0.f;
#pragma unroll
  for (int i = 0; i < 8; ++i) { x[i] = expf(x[i] - mx); sm += x[i]; }
  red[t] = sm; __syncthreads();
  for (int st = 128; st > 0; st >>= 1) {
    if (t < st) red[t] += red[t + st];
    __syncthreads();
  }
  const float rinv = 1.f / red[0];
#pragma unroll
  for (int i = 0; i < 8; ++i) S[(size_t)l * LVV + t + i * 256] = x[i] * rinv;
}

// ---- residual 1: src3 = srcbuf + (attn/txt branch) + pos ----
__global__ __launch_bounds__(256) void residual1_kernel(
    const float* __restrict__ srcbuf, const float* __restrict__ attnproj,
    const float* __restrict__ src2,   const float* __restrict__ posq,
    const float* __restrict__ post,   float* __restrict__ src3)
{
  const int row = blockIdx.x, t = threadIdx.x;
#pragma unroll
  for (int e = t; e < DD; e += 256) {
    const size_t idx = (size_t)row * DD + e;
    float add;
    if (row < LVV) add = attnproj[idx] + posq[idx];
    else           add = src2[idx] + post[(size_t)(row - LVV) * DD + e];
    src3[idx] = srcbuf[idx] + add;
  }
}

// ---- loss_sparsity: per-row sums of sigmoid(lw) excluding diagonal ----
__global__ __launch_bounds__(256) void loss_partial_kernel(
    const float* __restrict__ lw, float* __restrict__ rowsum)
{
  const int i = blockIdx.x, t = threadIdx.x;
  float s = 0.f;
#pragma unroll
  for (int c = 0; c < 8; ++c) {
    const int j = t + c * 256;
    if (j != i) s += fabsf(1.f / (1.f + expf(-lw[(size_t)i * LVV + j])));
  }
  __shared__ float red[256];
  red[t] = s; __syncthreads();
  for (int st = 128; st > 0; st >>= 1) {
    if (t < st) red[t] += red[t + st];
    __syncthreads();
  }
  if (t == 0) rowsum[i] = red[0];
}

__global__ void loss_final_kernel(const float* __restrict__ rowsum,
                                  float* __restrict__ out)
{
  float s = 0.f;
  for (int i = 0; i < LVV; ++i) s += rowsum[i];
  out[0] = s / ((float)LVV * (float)LVV);
}

// =====================================================================
extern "C" void kernel_launch(void* const* d_in, const int* in_sizes, int n_in,
                              void* d_out, int out_size, void* d_ws, size_t ws_size,
                              hipStream_t stream) {
  const float* src_vid = (const float*)d_in[0];
  const float* src_txt = (const float*)d_in[1];
  const float* ln1_w   = (const float*)d_in[2];
  const float* ln1_b   = (const float*)d_in[3];
  const float* wq = (const float*)d_in[4];  const float* bq = (const float*)d_in[5];
  const float* wk = (const float*)d_in[6];  const float* bk = (const float*)d_in[7];
  const float* wv = (const float*)d_in[8];  const float* bv = (const float*)d_in[9];
  const float* wo = (const float*)d_in[10]; const float* bo = (const float*)d_in[11];
  const float* lin1_w = (const float*)d_in[12]; const float* lin1_b = (const float*)d_in[13];
  const float* lin2_w = (const float*)d_in[14]; const float* lin2_b = (const float*)d_in[15];
  const float* ln2_w = (const float*)d_in[16]; const float* ln2_b = (const float*)d_in[17];
  const float* ln3_w = (const float*)d_in[18]; const float* ln3_b = (const float*)d_in[19];
  const float* lrw   = (const float*)d_in[20];
  float* out = (float*)d_out;

  // ---- workspace layout (floats) ----
  float* W = (float*)d_ws;
  float* srcbuf = W;                               // LL*DD
  float* src2   = srcbuf + (size_t)LL * DD;        // LL*DD
  float* src3   = src2   + (size_t)LL * DD;        // LL*DD
  float* posq   = src3   + (size_t)LL * DD;        // LVV*DD
  float* posk   = posq   + (size_t)LVV * DD;       // LVV*DD
  float* post   = posk   + (size_t)LVV * DD;       // LTT*DD
  float* q      = post   + (size_t)LTT * DD;       // LVV*DD
  float* k      = q      + (size_t)LVV * DD;       // LVV*DD
  float* Qm     = k      + (size_t)LVV * DD;       // LVV*DD
  float* Km     = Qm     + (size_t)LVV * DD;       // LVV*DD
  float* Vm     = Km     + (size_t)LVV * DD;       // LVV*DD
  float* S      = Vm     + (size_t)LVV * DD;       // LVV*LVV
  float* hidden = S      + (size_t)LVV * LVV;      // LL*FF
  float* rowsum = hidden + (size_t)LL * FF;        // LVV
  // reuse dead buffers:
  float* ctx      = k;       // after K projection, k is dead
  float* attnproj = q;       // after Q projection, q is dead
  float* src4     = srcbuf;  // after residual1, raw concat is dead
  float* src5     = src2;    // after residual1, LN1 output is dead

  const dim3 blk256(256), blkG(128);

  // 1) concat + positional encodings
  concat_kernel<<<LL, blk256, 0, stream>>>(src_vid, src_txt, srcbuf);
  xpos_kernel<<<LVV, blk256, 0, stream>>>(src_vid, posq, 0);
  xpos_kernel<<<LVV, blk256, 0, stream>>>(src_vid, posk, 1);
  sinepos_kernel<<<LTT, blk256, 0, stream>>>(post);

  // 2) LN1 and q/k build
  ln_kernel<<<LL, blk256, 0, stream>>>(srcbuf, ln1_w, ln1_b, src2);
  qk_build_kernel<<<LVV, blk256, 0, stream>>>(src2, posq, posk, q, k);

  // 3) Q/K/V projections (N x K weights => bKN=0); qh includes 1/sqrt(dk)
  gemm_wmma_f32<<<dim3(DD/64, LVV/64), blkG, 0, stream>>>(
      q, DD, 0, wq, DD, 0, 0, Qm, DD, 0, bq, nullptr, 0, 0, 0.125f, 0, DD);
  gemm_wmma_f32<<<dim3(DD/64, LVV/64), blkG, 0, stream>>>(
      k, DD, 0, wk, DD, 0, 0, Km, DD, 0, bk, nullptr, 0, 0, 1.0f, 0, DD);
  gemm_wmma_f32<<<dim3(DD/64, LVV/64), blkG, 0, stream>>>(
      src2, DD, 0, wv, DD, 0, 0, Vm, DD, 0, bv, nullptr, 0, 0, 1.0f, 0, DD);

  // 4) attention, head by head (score buffer S reused; stream-ordered)
  for (int h = 0; h < HH; ++h) {
    const int ho = h * DKK;
    // S = Qh . Kh^T   (both row-major over the dk contraction => bKN=0)
    gemm_wmma_f32<<<dim3(LVV/64, LVV/64), blkG, 0, stream>>>(
        Qm, DD, ho, Km, DD, ho, 0, S, LVV, 0,
        nullptr, nullptr, 0, 0, 1.0f, 0, DKK);
    // multiplicative retention mask + softmax (in place)
    mask_softmax_kernel<<<LVV, blk256, 0, stream>>>(S, lrw);
    // ctx_h = P . Vh   (V stored K x N => bKN=1)
    gemm_wmma_f32<<<dim3(DKK/64, LVV/64), blkG, 0, stream>>>(
        S, LVV, 0, Vm, DD, ho, 1, ctx, DD, ho,
        nullptr, nullptr, 0, 0, 1.0f, 0, LVV);
  }

  // 5) output projection
  gemm_wmma_f32<<<dim3(DD/64, LVV/64), blkG, 0, stream>>>(
      ctx, DD, 0, wo, DD, 0, 0, attnproj, DD, 0, bo, nullptr, 0, 0, 1.0f, 0, DD);

  // 6) residual + pos, LN2
  residual1_kernel<<<LL, blk256, 0, stream>>>(srcbuf, attnproj, src2, posq, post, src3);
  ln_kernel<<<LL, blk256, 0, stream>>>(src3, ln2_w, ln2_b, src4);

  // 7) FFN: GELU(src4 @ lin1^T + b1) @ lin2^T + b2, + residual src3
  gemm_wmma_f32<<<dim3(FF/64, LL/64), blkG, 0, stream>>>(
      src4, DD, 0, lin1_w, DD, 0, 0, hidden, FF, 0,
      lin1_b, nullptr, 0, 0, 1.0f, 1, DD);
  gemm_wmma_f32<<<dim3(DD/64, LL/64), blkG, 0, stream>>>(
      hidden, FF, 0, lin2_w, FF, 0, 0, src5, DD, 0,
      lin2_b, src3, DD, 0, 1.0f, 0, FF);

  // 8) LN3 straight into d_out (vid rows then txt rows are contiguous)
  ln_kernel<<<LL, blk256, 0, stream>>>(src5, ln3_w, ln3_b, out);

  // 9) sparsity loss -> out[LL*DD] (deterministic two-stage reduction)
  loss_partial_kernel<<<LVV, blk256, 0, stream>>>(lrw, rowsum);
  loss_final_kernel<<<1, 1, 0, stream>>>(rowsum, out + (size_t)LL * DD);
}